// GatedGCN_1d_77343771066508
// MI455X (gfx1250) — compile-verified
//
#include <hip/hip_runtime.h>
#include <cstddef>

#define D 64
#define BN_EPS 1e-5f
#define DEG_EPS 1e-6f

typedef float v2f __attribute__((ext_vector_type(2)));
typedef float v8f __attribute__((ext_vector_type(8)));

// ---------------------------------------------------------------------------
// Generic D=64 GEMM:  O[r][n] = sum_k X[r][k] * W[n][k] + bias[n]
// One wave computes a 16-row x 64-col output stripe using V_WMMA_F32_16X16X4_F32.
// A-layout (ISA 7.12.2, 32-bit A 16x4): lane lid (0..15) = row M, half = lane>>4,
//   a[j] holds K = 4*kstep + 2*half + j  -> one aligned float2 load per kstep.
// B-layout (4x16, row striped across lanes): b[j] = W[N=lid][K=4*kstep+2*half+j].
// C/D layout: c[v] = O[M = v + 8*half][N = lid].
// ---------------------------------------------------------------------------
__global__ __launch_bounds__(128)
void gemm64_wmma(const float* __restrict__ X, const float* __restrict__ W,
                 const float* __restrict__ bias, float* __restrict__ O, int nrows)
{
    const int tid   = threadIdx.x;
    const int wave  = tid >> 5;
    const int lane  = tid & 31;
    const int half_ = lane >> 4;
    const int lid   = lane & 15;
    const int rowbase = (blockIdx.x * 4 + wave) * 16;
    if (rowbase >= nrows) return;              // wave-uniform guard (N % 16 == 0)

    const float* arow = X + (size_t)(rowbase + lid) * D + 2 * half_;
    v2f a[16];
#pragma unroll
    for (int k = 0; k < 16; ++k) a[k] = *(const v2f*)(arow + 4 * k);

#pragma unroll
    for (int nt = 0; nt < 4; ++nt) {
        const float* wrow = W + (size_t)(nt * 16 + lid) * D + 2 * half_;
        v8f c = {0.f, 0.f, 0.f, 0.f, 0.f, 0.f, 0.f, 0.f};
#pragma unroll
        for (int k = 0; k < 16; ++k) {
            v2f b = *(const v2f*)(wrow + 4 * k);
            c = __builtin_amdgcn_wmma_f32_16x16x4_f32(false, a[k], false, b,
                                                      (short)0, c, false, false);
        }
        const float bv = bias[nt * 16 + lid];
        float* obase = O + (size_t)rowbase * D + nt * 16 + lid;
#pragma unroll
        for (int v = 0; v < 8; ++v)
            obase[(size_t)(v + 8 * half_) * D] = c[v] + bv;
    }
}

// ---------------------------------------------------------------------------
// Fused edge kernel: t = (e @ B3w^T + B3b) + B1h[src] + B2h[dst]
// Also accumulates per-feature sum / sumsq for edge BatchNorm
// (LDS ds_add_f32 reduction, then 1 global atomic per feature per block).
// ---------------------------------------------------------------------------
__global__ __launch_bounds__(128)
void edge_fused(const float* __restrict__ e, const float* __restrict__ W,
                const float* __restrict__ bias,
                const float* __restrict__ B1h, const float* __restrict__ B2h,
                const int* __restrict__ src, const int* __restrict__ dst,
                float* __restrict__ t, float* __restrict__ statsE, int E)
{
    __shared__ float lsum[D];
    __shared__ float lsq[D];
    const int tid = threadIdx.x;
    if (tid < D) { lsum[tid] = 0.f; lsq[tid] = 0.f; }
    __syncthreads();

    const int wave  = tid >> 5;
    const int lane  = tid & 31;
    const int half_ = lane >> 4;
    const int lid   = lane & 15;
    const int ebase = (blockIdx.x * 4 + wave) * 16;
    const bool active = (ebase + 16) <= E;     // E % 64 == 0 in practice

    if (active) {
        const float* arow = e + (size_t)(ebase + lid) * D + 2 * half_;
        v2f a[16];
#pragma unroll
        for (int k = 0; k < 16; ++k) a[k] = *(const v2f*)(arow + 4 * k);

#pragma unroll
        for (int nt = 0; nt < 4; ++nt) {
            const int f = nt * 16 + lid;
            const float* wrow = W + (size_t)f * D + 2 * half_;
            v8f c = {0.f, 0.f, 0.f, 0.f, 0.f, 0.f, 0.f, 0.f};
#pragma unroll
            for (int k = 0; k < 16; ++k) {
                v2f b = *(const v2f*)(wrow + 4 * k);
                c = __builtin_amdgcn_wmma_f32_16x16x4_f32(false, a[k], false, b,
                                                          (short)0, c, false, false);
            }
            const float bv = bias[f];
            float fsum = 0.f, fsq = 0.f;
#pragma unroll
            for (int v = 0; v < 8; ++v) {
                const int edge = ebase + v + 8 * half_;
                const int s  = src[edge];
                const int d2 = dst[edge];
                float val = c[v] + bv
                          + B1h[(size_t)s  * D + f]
                          + B2h[(size_t)d2 * D + f];
                t[(size_t)edge * D + f] = val;
                fsum += val;
                fsq  += val * val;
            }
            atomicAdd(&lsum[f], fsum);
            atomicAdd(&lsq[f],  fsq);
        }
    }
    __syncthreads();
    if (tid < D) {
        atomicAdd(&statsE[tid],     lsum[tid]);
        atomicAdd(&statsE[D + tid], lsq[tid]);
    }
}

// stats[0..63]=sum, [64..127]=sumsq  ->  [128..191]=mean, [192..255]=rsqrt(var+eps)
__global__ void bn_reduce(float* __restrict__ stats, float count)
{
    const int f = threadIdx.x;
    if (f < D) {
        float m = stats[f] / count;
        float v = stats[D + f] / count - m * m;
        stats[128 + f] = m;
        stats[192 + f] = rsqrtf(v + BN_EPS);
    }
}

// ---------------------------------------------------------------------------
// Edge pass 2: sigma = sigmoid(relu(BN(t)) + e); scatter-add 4 accumulators;
// also writes e (unchanged) into the output buffer.
// ---------------------------------------------------------------------------
__global__ __launch_bounds__(256)
void edge_pass2(const float* __restrict__ t, const float* __restrict__ e,
                const int* __restrict__ src, const int* __restrict__ dst,
                const float* __restrict__ A2h, const float* __restrict__ A3h,
                const float* __restrict__ statsE,
                const float* __restrict__ gamma, const float* __restrict__ beta,
                float* __restrict__ num_f, float* __restrict__ den_f,
                float* __restrict__ num_b, float* __restrict__ den_b,
                float* __restrict__ out_e, int total)
{
    const int idx = blockIdx.x * 256 + threadIdx.x;
    if (idx >= total) return;
    const int f    = idx & (D - 1);
    const int edge = idx >> 6;

    const float ev   = e[idx];
    float g = gamma[f] * (t[idx] - statsE[128 + f]) * statsE[192 + f] + beta[f];
    g = fmaxf(g, 0.f) + ev;
    const float sig = 1.f / (1.f + __expf(-g));

    const int s  = src[edge];
    const int d2 = dst[edge];
    const float a2 = A2h[(size_t)s  * D + f];
    const float a3 = A3h[(size_t)d2 * D + f];

    atomicAdd(&num_f[(size_t)d2 * D + f], sig * a2);
    atomicAdd(&den_f[(size_t)d2 * D + f], sig);
    atomicAdd(&num_b[(size_t)s  * D + f], sig * a3);
    atomicAdd(&den_b[(size_t)s  * D + f], sig);

    out_e[idx] = ev;
}

// ---------------------------------------------------------------------------
// Node pass 1: pre = A1h + num_f/(den_f+eps) + num_b/(den_b+eps)  (in place over
// A1h) + node-BN statistics via LDS reduction.
// ---------------------------------------------------------------------------
__global__ __launch_bounds__(256)
void node_pre_stats(float* __restrict__ pre /* = A1h in/out */,
                    const float* __restrict__ num_f, const float* __restrict__ den_f,
                    const float* __restrict__ num_b, const float* __restrict__ den_b,
                    float* __restrict__ statsH, int total)
{
    __shared__ float lsum[D];
    __shared__ float lsq[D];
    const int tid = threadIdx.x;
    if (tid < D) { lsum[tid] = 0.f; lsq[tid] = 0.f; }
    __syncthreads();

    const int idx = blockIdx.x * 256 + tid;
    if (idx < total) {
        const int f = idx & (D - 1);
        float val = pre[idx]
                  + num_f[idx] / (den_f[idx] + DEG_EPS)
                  + num_b[idx] / (den_b[idx] + DEG_EPS);
        pre[idx] = val;
        atomicAdd(&lsum[f], val);
        atomicAdd(&lsq[f],  val * val);
    }
    __syncthreads();
    if (tid < D) {
        atomicAdd(&statsH[tid],     lsum[tid]);
        atomicAdd(&statsH[D + tid], lsq[tid]);
    }
}

__global__ __launch_bounds__(256)
void node_pass2(const float* __restrict__ pre, const float* __restrict__ h,
                const float* __restrict__ gamma, const float* __restrict__ beta,
                const float* __restrict__ statsH, float* __restrict__ outh, int total)
{
    const int idx = blockIdx.x * 256 + threadIdx.x;
    if (idx >= total) return;
    const int f = idx & (D - 1);
    float g = gamma[f] * (pre[idx] - statsH[128 + f]) * statsH[192 + f] + beta[f];
    outh[idx] = fmaxf(g, 0.f) + h[idx];
}

// ---------------------------------------------------------------------------
extern "C" void kernel_launch(void* const* d_in, const int* in_sizes, int n_in,
                              void* d_out, int out_size, void* d_ws, size_t ws_size,
                              hipStream_t stream)
{
    const float* h   = (const float*)d_in[0];
    const float* e   = (const float*)d_in[1];
    const int*   src = (const int*)d_in[2];
    const int*   dst = (const int*)d_in[3];
    const float* A1w = (const float*)d_in[4],  *A1b = (const float*)d_in[5];
    const float* A2w = (const float*)d_in[6],  *A2b = (const float*)d_in[7];
    const float* A3w = (const float*)d_in[8],  *A3b = (const float*)d_in[9];
    const float* B1w = (const float*)d_in[10], *B1b = (const float*)d_in[11];
    const float* B2w = (const float*)d_in[12], *B2b = (const float*)d_in[13];
    const float* B3w = (const float*)d_in[14], *B3b = (const float*)d_in[15];
    const float* bnhg = (const float*)d_in[16], *bnhb = (const float*)d_in[17];
    const float* bneg = (const float*)d_in[18], *bneb = (const float*)d_in[19];

    const int N = in_sizes[0] / D;     // 50000
    const int E = in_sizes[1] / D;     // 800000
    const size_t nodeElems = (size_t)N * D;
    const size_t edgeElems = (size_t)E * D;

    float* ws = (float*)d_ws;
    float* A1h  = ws; ws += nodeElems;   // later reused as "pre"
    float* A2h  = ws; ws += nodeElems;
    float* A3h  = ws; ws += nodeElems;
    float* B1h  = ws; ws += nodeElems;
    float* B2h  = ws; ws += nodeElems;
    float* tg   = ws; ws += edgeElems;   // pre-BN edge gate
    float* numf = ws; ws += nodeElems;   // numf..denb contiguous (one memset)
    float* denf = ws; ws += nodeElems;
    float* numb = ws; ws += nodeElems;
    float* denb = ws; ws += nodeElems;
    float* statsE = ws; ws += 256;
    float* statsH = ws; ws += 256;       // statsE/statsH contiguous

    // Zero the accumulators every call (graph-capturable memset nodes).
    hipMemsetAsync(numf,   0, nodeElems * 4 * sizeof(float), stream);
    hipMemsetAsync(statsE, 0, 512 * sizeof(float), stream);

    float* out_h = (float*)d_out;
    float* out_e = (float*)d_out + nodeElems;

    const int gemmBlocksN = (N + 63) / 64;
    gemm64_wmma<<<gemmBlocksN, 128, 0, stream>>>(h, A1w, A1b, A1h, N);
    gemm64_wmma<<<gemmBlocksN, 128, 0, stream>>>(h, A2w, A2b, A2h, N);
    gemm64_wmma<<<gemmBlocksN, 128, 0, stream>>>(h, A3w, A3b, A3h, N);
    gemm64_wmma<<<gemmBlocksN, 128, 0, stream>>>(h, B1w, B1b, B1h, N);
    gemm64_wmma<<<gemmBlocksN, 128, 0, stream>>>(h, B2w, B2b, B2h, N);

    edge_fused<<<(E + 63) / 64, 128, 0, stream>>>(e, B3w, B3b, B1h, B2h,
                                                  src, dst, tg, statsE, E);
    bn_reduce<<<1, 64, 0, stream>>>(statsE, (float)E);

    edge_pass2<<<(int)((edgeElems + 255) / 256), 256, 0, stream>>>(
        tg, e, src, dst, A2h, A3h, statsE, bneg, bneb,
        numf, denf, numb, denb, out_e, (int)edgeElems);

    node_pre_stats<<<(int)((nodeElems + 255) / 256), 256, 0, stream>>>(
        A1h, numf, denf, numb, denb, statsH, (int)nodeElems);
    bn_reduce<<<1, 64, 0, stream>>>(statsH, (float)N);

    node_pass2<<<(int)((nodeElems + 255) / 256), 256, 0, stream>>>(
        A1h, h, bnhg, bnhb, statsH, out_h, (int)nodeElems);
}